// DLinear_21131239096346
// MI455X (gfx1250) — compile-verified
//
#include <hip/hip_runtime.h>

typedef __attribute__((ext_vector_type(2))) float v2f;
typedef __attribute__((ext_vector_type(4))) float v4f;
typedef __attribute__((ext_vector_type(8))) float v8f;

#define Bsz   2048
#define Lsz   512
#define Csz   158
#define Ksz   (Lsz * Csz)      // 80896 = flattened reduction length
#define KERW  25
#define PADW  12               // (KERW-1)/2
#define SPLIT 16               // K splits per row-group
#define WPB   8                // waves per block (256 threads)
#define KPB   (Ksz / SPLIT)    // 5056 K per block
#define KPW   (KPB / WPB)      // 632 K per wave (multiple of 8)
#define NITER (KPW / 8)        // 79 macro-iterations (compile-time constant)

// ---------------------------------------------------------------------------
// Kernel 1: fold moving-average operator + decoder into one weight vector.
// W2t[l*C + c] = w_dec[c] * ( w_s[c,l] + (1/25) * sum_j (w_t-w_s)[c,j]*M[j,l] )
// M columns: interior l -> plain window; l==0 / l==L-1 -> replicate-pad
// triangular weights.
// ---------------------------------------------------------------------------
__global__ void precompute_w2(const float* __restrict__ wS,
                              const float* __restrict__ wT,
                              const float* __restrict__ wD,
                              float* __restrict__ W2t) {
  int idx = blockIdx.x * blockDim.x + threadIdx.x;
  if (idx >= Ksz) return;
  int l = idx / Csz;
  int c = idx % Csz;
  const float* s = wS + c * Lsz;
  const float* t = wT + c * Lsz;
  float acc = 0.0f;
  if (l == 0) {
    for (int j = 0; j <= PADW; ++j)
      acc += (t[j] - s[j]) * (float)(PADW + 1 - j);
  } else if (l == Lsz - 1) {
    for (int j = Lsz - 1 - PADW; j < Lsz; ++j)
      acc += (t[j] - s[j]) * (float)(PADW + 1 - (Lsz - 1 - j));
  } else {
    int lo = l - PADW; if (lo < 0) lo = 0;
    int hi = l + PADW; if (hi > Lsz - 1) hi = Lsz - 1;
    for (int j = lo; j <= hi; ++j)
      acc += (t[j] - s[j]);
  }
  float weff = s[l] + acc * (1.0f / (float)KERW);
  W2t[idx] = wD[c] * weff;
}

// ---------------------------------------------------------------------------
// Kernel 2: const = b_dec + sum_c w_dec[c]*(b_s[c]+b_t[c])
// ---------------------------------------------------------------------------
__global__ void bias_const(const float* __restrict__ bS,
                           const float* __restrict__ bT,
                           const float* __restrict__ wD,
                           const float* __restrict__ bD,
                           float* __restrict__ cst) {
  __shared__ float red[256];
  int t = threadIdx.x;
  float v = 0.0f;
  if (t < Csz) v = wD[t] * (bS[t] + bT[t]);
  red[t] = v;
  __syncthreads();
  for (int o = 128; o > 0; o >>= 1) {
    if (t < o) red[t] += red[t + o];
    __syncthreads();
  }
  if (t == 0) *cst = red[0] + bD[0];
}

// ---------------------------------------------------------------------------
// Kernel 3 (main, WMMA): partial[b, split] = sum over K-slice of
// x[b, k] * W2t[k], computed with V_WMMA_F32_16X16X4_F32.
//   A (16x4 f32): lane m(0-15) holds x[row m, g+0..1], lane m+16 holds
//                 x[row m, g+4..5] (vgpr pair from one b128 load)
//   B (4x16 f32): all lanes load the same W2 chunk for their lane-group ->
//                 every C column equals the desired dot; read column 0.
// Loop trip count is a compile-time constant and pointers are bumped, so the
// loop stays uniform (scalar counter) with minimal per-iteration VALU.
// ---------------------------------------------------------------------------
__global__ void __launch_bounds__(256, 1)
main_dot(const float* __restrict__ x,
         const float* __restrict__ W2t,
         float* __restrict__ partial) {
  __shared__ float red[WPB * 16];
  const int rg   = blockIdx.x;          // 0..127 row-group (16 rows)
  const int sp   = blockIdx.y;          // 0..SPLIT-1 K split
  const int wave = threadIdx.x >> 5;
  const int lane = threadIdx.x & 31;
  const int m    = lane & 15;
  const int hi   = lane >> 4;           // 0: K slots 0,1   1: K slots 2,3

  const int k0 = sp * KPB + wave * KPW + 4 * hi;
  const float* xp = x + (size_t)(rg * 16 + m) * (size_t)Ksz + k0;
  const float* wp = W2t + k0;

  v8f acc0 = {};
  v8f acc1 = {};
#pragma unroll 4
  for (int it = 0; it < NITER; ++it) {
    // 16B per lane: lanes0-15 cover K=g..g+3, lanes16-31 cover K=g+4..g+7
    v4f a = *(const v4f*)xp;
    v4f w = *(const v4f*)wp;             // broadcast into all 16 cols
    xp += 8;
    wp += 8;
    v2f a01 = {a.x, a.y}, a23 = {a.z, a.w};
    v2f w01 = {w.x, w.y}, w23 = {w.z, w.w};
    acc0 = __builtin_amdgcn_wmma_f32_16x16x4_f32(
        false, a01, false, w01, (short)0, acc0, false, false);
    acc1 = __builtin_amdgcn_wmma_f32_16x16x4_f32(
        false, a23, false, w23, (short)0, acc1, false, false);
  }
  v8f acc = acc0 + acc1;

  // C layout: vgpr r -> lane0 = C[M=r, N=0], lane16 = C[M=r+8, N=0]
  if (lane == 0) {
#pragma unroll
    for (int r = 0; r < 8; ++r) red[wave * 16 + r] = acc[r];
  } else if (lane == 16) {
#pragma unroll
    for (int r = 0; r < 8; ++r) red[wave * 16 + 8 + r] = acc[r];
  }
  __syncthreads();
  if (threadIdx.x < 16) {
    float p = 0.0f;
#pragma unroll
    for (int w = 0; w < WPB; ++w) p += red[w * 16 + threadIdx.x];
    partial[(size_t)(rg * 16 + threadIdx.x) * SPLIT + sp] = p;
  }
}

// ---------------------------------------------------------------------------
// Kernel 4: fixed-order reduction over K-splits + bias (deterministic).
// ---------------------------------------------------------------------------
__global__ void finalize(const float* __restrict__ partial,
                         const float* __restrict__ cst,
                         float* __restrict__ out) {
  int b = blockIdx.x * blockDim.x + threadIdx.x;
  if (b >= Bsz) return;
  float v = *cst;
#pragma unroll
  for (int s = 0; s < SPLIT; ++s) v += partial[(size_t)b * SPLIT + s];
  out[b] = v;
}

extern "C" void kernel_launch(void* const* d_in, const int* in_sizes, int n_in,
                              void* d_out, int out_size, void* d_ws, size_t ws_size,
                              hipStream_t stream) {
  (void)in_sizes; (void)n_in; (void)out_size; (void)ws_size;
  const float* x  = (const float*)d_in[0];
  const float* wS = (const float*)d_in[1];
  const float* bS = (const float*)d_in[2];
  const float* wT = (const float*)d_in[3];
  const float* bT = (const float*)d_in[4];
  const float* wD = (const float*)d_in[5];
  const float* bD = (const float*)d_in[6];
  float* out = (float*)d_out;

  // workspace layout: [W2t: Ksz][cst: 1][partial: Bsz*SPLIT]  (~455 KB)
  float* W2t     = (float*)d_ws;
  float* cst     = W2t + Ksz;
  float* partial = cst + 1;

  precompute_w2<<<(Ksz + 255) / 256, 256, 0, stream>>>(wS, wT, wD, W2t);
  bias_const<<<1, 256, 0, stream>>>(bS, bT, wD, bD, cst);
  dim3 grid(Bsz / 16, SPLIT);
  main_dot<<<grid, 256, 0, stream>>>(x, W2t, partial);
  finalize<<<(Bsz + 255) / 256, 256, 0, stream>>>(partial, cst, out);
}